// GNN_42331197670193
// MI455X (gfx1250) — compile-verified
//
#include <hip/hip_runtime.h>
#include <hip/hip_bf16.h>

#define N_NODES 1001112
#define N_EDGES 4000000
#define HID 64
#define N_GRAPHS 512

typedef __attribute__((ext_vector_type(2))) float v2f;
typedef __attribute__((ext_vector_type(8))) float v8f;

// ---------------- degree / norm ----------------

__global__ __launch_bounds__(256) void init_deg_kernel(float* __restrict__ deg) {
    int i = blockIdx.x * 256 + threadIdx.x;
    if (i < N_NODES) deg[i] = 1.0f;  // self-loop contribution
}

__global__ __launch_bounds__(256) void deg_edges_kernel(const long long* __restrict__ ei,
                                                        float* __restrict__ deg) {
    long long e = (long long)blockIdx.x * 256 + threadIdx.x;
    if (e < N_EDGES) {
        long long d = ei[N_EDGES + e];
        atomicAdd(deg + d, 1.0f);
    }
}

__global__ __launch_bounds__(256) void rsqrt_kernel(float* __restrict__ deg) {
    int i = blockIdx.x * 256 + threadIdx.x;
    if (i < N_NODES) deg[i] = rsqrtf(deg[i]);
}

// ---------------- layer 1: x @ W1  ([N,3]@[3,64]) ----------------

__global__ __launch_bounds__(256) void xw1_kernel(const float* __restrict__ x,
                                                  const float* __restrict__ W1,
                                                  float* __restrict__ out) {
    __shared__ float w[3 * 64];
    if (threadIdx.x < 192) w[threadIdx.x] = W1[threadIdx.x];
    __syncthreads();
    long long tid = (long long)blockIdx.x * 256 + threadIdx.x;
    long long i = tid >> 4;
    int part = (int)(tid & 15);
    if (i >= N_NODES) return;
    float x0 = x[i * 3 + 0];
    float x1 = x[i * 3 + 1];
    float x2 = x[i * 3 + 2];
    int f = part * 4;
    float4 r;
    r.x = x0 * w[f + 0] + x1 * w[64 + f + 0] + x2 * w[128 + f + 0];
    r.y = x0 * w[f + 1] + x1 * w[64 + f + 1] + x2 * w[128 + f + 1];
    r.z = x0 * w[f + 2] + x1 * w[64 + f + 2] + x2 * w[128 + f + 2];
    r.w = x0 * w[f + 3] + x1 * w[64 + f + 3] + x2 * w[128 + f + 3];
    *reinterpret_cast<float4*>(out + i * 64 + f) = r;
}

// ---------------- message passing ----------------

// acc[i,:] = xw[i,:] * dinv[i]^2   (self-loop term; also initializes acc)
__global__ __launch_bounds__(256) void selfloop_kernel(const float* __restrict__ xw,
                                                       const float* __restrict__ dinv,
                                                       float* __restrict__ acc) {
    long long tid = (long long)blockIdx.x * 256 + threadIdx.x;
    long long i = tid >> 4;
    int part = (int)(tid & 15);
    if (i >= N_NODES) return;
    float di = dinv[i];
    float s = di * di;
    float4 v = *reinterpret_cast<const float4*>(xw + i * 64 + part * 4);
    v.x *= s; v.y *= s; v.z *= s; v.w *= s;
    *reinterpret_cast<float4*>(acc + i * 64 + part * 4) = v;
}

// 16 lanes per edge; coalesced float4 gather of the source row, atomic scatter to dst.
__global__ __launch_bounds__(256) void edge_scatter_kernel(const long long* __restrict__ ei,
                                                           const float* __restrict__ dinv,
                                                           const float* __restrict__ xw,
                                                           float* __restrict__ acc) {
    long long tid = (long long)blockIdx.x * 256 + threadIdx.x;
    long long e = tid >> 4;
    int part = (int)(tid & 15);
    if (e >= N_EDGES) return;
    long long s = ei[e];
    long long d = ei[N_EDGES + e];
    float nrm = dinv[s] * dinv[d];
    float4 v = *reinterpret_cast<const float4*>(xw + s * 64 + part * 4);
    float* ap = acc + d * 64 + part * 4;
    atomicAdd(ap + 0, v.x * nrm);
    atomicAdd(ap + 1, v.y * nrm);
    atomicAdd(ap + 2, v.z * nrm);
    atomicAdd(ap + 3, v.w * nrm);
}

// h = relu(acc + b)   (may run in place)
__global__ __launch_bounds__(256) void bias_relu_kernel(const float* __restrict__ acc,
                                                        const float* __restrict__ b,
                                                        float* __restrict__ h) {
    long long tid = (long long)blockIdx.x * 256 + threadIdx.x;
    long long i = tid >> 4;
    int part = (int)(tid & 15);
    if (i >= N_NODES) return;
    int f = part * 4;
    float4 v = *reinterpret_cast<const float4*>(acc + i * 64 + f);
    float4 bb = *reinterpret_cast<const float4*>(b + f);
    v.x = fmaxf(v.x + bb.x, 0.0f);
    v.y = fmaxf(v.y + bb.y, 0.0f);
    v.z = fmaxf(v.z + bb.z, 0.0f);
    v.w = fmaxf(v.w + bb.w, 0.0f);
    *reinterpret_cast<float4*>(h + i * 64 + f) = v;
}

// ---------------- layer 2 GEMM: [N,64] @ [64,64] via V_WMMA_F32_16X16X4_F32 ----------------
// Each wave computes a 16x64 output slab: 4 accumulator tiles, 16 K-steps of K=4.
// A 16x4 f32 fragment: lane<16 -> row=lane, (K0,K1); lane>=16 -> row=lane-16, (K2,K3).
// B 4x16 f32 fragment: lane<16 -> col=lane, rows (K0,K1); lane>=16 -> col=lane-16, (K2,K3).
// D 16x16 f32: VGPR g: lane<16 -> M=g,N=lane; lane>=16 -> M=8+g,N=lane-16.
//
// W is async-staged raw into LDS, then pre-swizzled so every B fragment is one
// contiguous, aligned ds_load_b64:  w_swz[p*64 + c] = { W[2p][c], W[2p+1][c] }.
// Partial last tile: A row index is clamped (garbage only pollutes its own D row,
// which the store-side guard drops) -> branch-free hot loop.
__global__ __launch_bounds__(256) void gemm64_wmma_kernel(const float* __restrict__ H,
                                                          const float* __restrict__ W,
                                                          float* __restrict__ out) {
    __shared__ float w_raw[64 * 64];   // 16 KB raw copy of W
    __shared__ v2f   w_swz[32 * 64];   // 16 KB K-pair-swizzled copy

    // Stage W (16 KB) into LDS with gfx1250 async-to-LDS loads (ASYNCcnt-tracked).
    {
        unsigned lds_base = (unsigned)(size_t)(void*)&w_raw[0];
#pragma unroll
        for (int j = 0; j < 4; ++j) {
            unsigned byteoff = (threadIdx.x + j * 256u) * 16u;
            unsigned ldsaddr = lds_base + byteoff;
            const char* gaddr = (const char*)W + byteoff;
            asm volatile("global_load_async_to_lds_b128 %0, %1, off"
                         :: "v"(ldsaddr), "v"(gaddr)
                         : "memory");
        }
        asm volatile("s_wait_asynccnt 0x0" ::: "memory");
    }
    __syncthreads();

    // Swizzle into K-pair layout (one-time, per block).
#pragma unroll
    for (int j = 0; j < 8; ++j) {
        int idx = threadIdx.x + j * 256;   // idx = p*64 + c, p in [0,32), c in [0,64)
        int p = idx >> 6;
        int c = idx & 63;
        v2f t;
        t.x = w_raw[(2 * p) * 64 + c];
        t.y = w_raw[(2 * p + 1) * 64 + c];
        w_swz[idx] = t;
    }
    __syncthreads();

    int wave = threadIdx.x >> 5;               // 0..7
    int lane = threadIdx.x & 31;
    long long tile = (long long)blockIdx.x * 8 + wave;
    long long m0 = tile * 16;
    if (m0 >= N_NODES) return;                 // wave-uniform exit

    int r = lane & 15;
    int khalf2 = (lane >> 4);                  // 0 or 1 -> K offset 0 or 2 (pair index offset)
    long long row = m0 + r;
    long long row_c = (row < N_NODES) ? row : (long long)(N_NODES - 1);
    const v2f* __restrict__ hrow = reinterpret_cast<const v2f*>(H + row_c * 64);

    v8f acc0 = {}, acc1 = {}, acc2 = {}, acc3 = {};

#pragma unroll 4
    for (int kk = 0; kk < 16; ++kk) {
        v2f a = hrow[kk * 2 + khalf2];         // one global_load_b64, branch-free

        int p = kk * 2 + khalf2;               // K-pair index for this lane half
        const v2f* bp = &w_swz[p * 64 + r];
        v2f b0 = bp[0];                        // single aligned ds_load_b64 each,
        v2f b1 = bp[16];                       // lands directly in a VGPR pair
        v2f b2 = bp[32];
        v2f b3 = bp[48];

        acc0 = __builtin_amdgcn_wmma_f32_16x16x4_f32(false, a, false, b0, (short)0, acc0, false, false);
        acc1 = __builtin_amdgcn_wmma_f32_16x16x4_f32(false, a, false, b1, (short)0, acc1, false, false);
        acc2 = __builtin_amdgcn_wmma_f32_16x16x4_f32(false, a, false, b2, (short)0, acc2, false, false);
        acc3 = __builtin_amdgcn_wmma_f32_16x16x4_f32(false, a, false, b3, (short)0, acc3, false, false);
    }

    long long mbase = m0 + (long long)(lane >> 4) * 8;
    int n = r;
#pragma unroll
    for (int g = 0; g < 8; ++g) {
        long long m = mbase + g;
        if (m < N_NODES) {
            out[m * 64 +  0 + n] = acc0[g];
            out[m * 64 + 16 + n] = acc1[g];
            out[m * 64 + 32 + n] = acc2[g];
            out[m * 64 + 48 + n] = acc3[g];
        }
    }
}

// ---------------- heads ----------------

// logits[i] = mask ? h[i,:]·Wp + bp : -1e9   (one wave per node)
__global__ __launch_bounds__(256) void logits_kernel(const float* __restrict__ h,
                                                     const float* __restrict__ Wp,
                                                     const float* __restrict__ bp,
                                                     const long long* __restrict__ mask,
                                                     float* __restrict__ out) {
    int lane = threadIdx.x & 31;
    long long i = ((long long)blockIdx.x * 256 + threadIdx.x) >> 5;
    if (i >= N_NODES) return;
    float2 hv = *reinterpret_cast<const float2*>(h + i * 64 + lane * 2);
    float2 wv = *reinterpret_cast<const float2*>(Wp + lane * 2);
    float p = hv.x * wv.x + hv.y * wv.y;
    for (int off = 16; off >= 1; off >>= 1) p += __shfl_xor(p, off, 32);
    if (lane == 0) out[i] = (mask[i] == 0) ? -1e9f : (p + bp[0]);
}

__global__ __launch_bounds__(256) void pool_zero_kernel(float* __restrict__ sums,
                                                        float* __restrict__ cnts) {
    int i = blockIdx.x * 256 + threadIdx.x;
    if (i < N_GRAPHS * 64) sums[i] = 0.0f;
    if (i < N_GRAPHS) cnts[i] = 0.0f;
}

__global__ __launch_bounds__(256) void pool_accum_kernel(const float* __restrict__ h,
                                                         const long long* __restrict__ batch,
                                                         float* __restrict__ sums,
                                                         float* __restrict__ cnts) {
    long long tid = (long long)blockIdx.x * 256 + threadIdx.x;
    long long i = tid >> 4;
    int part = (int)(tid & 15);
    if (i >= N_NODES) return;
    long long g = batch[i];
    float4 v = *reinterpret_cast<const float4*>(h + i * 64 + part * 4);
    float* sp = sums + g * 64 + part * 4;
    atomicAdd(sp + 0, v.x);
    atomicAdd(sp + 1, v.y);
    atomicAdd(sp + 2, v.z);
    atomicAdd(sp + 3, v.w);
    if (part == 0) atomicAdd(cnts + g, 1.0f);
}

// v[g] = tanh(mean(h over graph)·Wv + bv)   (one wave per graph)
__global__ __launch_bounds__(256) void value_kernel(const float* __restrict__ sums,
                                                    const float* __restrict__ cnts,
                                                    const float* __restrict__ Wv,
                                                    const float* __restrict__ bv,
                                                    float* __restrict__ out) {
    int lane = threadIdx.x & 31;
    int g = (blockIdx.x * 256 + threadIdx.x) >> 5;
    if (g >= N_GRAPHS) return;
    float c = fmaxf(cnts[g], 1.0f);
    float2 sv = *reinterpret_cast<const float2*>(sums + g * 64 + lane * 2);
    float2 wv = *reinterpret_cast<const float2*>(Wv + lane * 2);
    float p = (sv.x * wv.x + sv.y * wv.y);
    for (int off = 16; off >= 1; off >>= 1) p += __shfl_xor(p, off, 32);
    if (lane == 0) out[g] = tanhf(p / c + bv[0]);
}

// ---------------- launcher ----------------

extern "C" void kernel_launch(void* const* d_in, const int* in_sizes, int n_in,
                              void* d_out, int out_size, void* d_ws, size_t ws_size,
                              hipStream_t stream) {
    const float*     x    = (const float*)d_in[0];
    const long long* ei   = (const long long*)d_in[1];
    const long long* bat  = (const long long*)d_in[2];
    const long long* msk  = (const long long*)d_in[3];
    const float*     W1   = (const float*)d_in[4];
    const float*     b1   = (const float*)d_in[5];
    const float*     W2   = (const float*)d_in[6];
    const float*     b2   = (const float*)d_in[7];
    const float*     Wp   = (const float*)d_in[8];
    const float*     bp   = (const float*)d_in[9];
    const float*     Wv   = (const float*)d_in[10];
    const float*     bv   = (const float*)d_in[11];

    float* logits = (float*)d_out;            // [N_NODES]
    float* vout   = logits + N_NODES;         // [N_GRAPHS]

    // workspace layout
    float* dinv = (float*)d_ws;                          // [N]
    float* B1   = dinv + N_NODES;                        // [N*64]
    float* B2   = B1 + (size_t)N_NODES * 64;             // [N*64]
    float* sums = B2 + (size_t)N_NODES * 64;             // [G*64]
    float* cnts = sums + (size_t)N_GRAPHS * 64;          // [G]

    const int TB = 256;
    int gN    = (N_NODES + TB - 1) / TB;                        // per-node, 1 thread each
    int gE    = (N_EDGES + TB - 1) / TB;                        // per-edge, 1 thread each
    int gN16  = (int)(((long long)N_NODES * 16 + TB - 1) / TB); // per-node, 16 lanes
    int gE16  = (int)(((long long)N_EDGES * 16 + TB - 1) / TB); // per-edge, 16 lanes
    int gN32  = (int)(((long long)N_NODES * 32 + TB - 1) / TB); // per-node, 1 wave
    int tiles = (N_NODES + 15) / 16;
    int gGemm = (tiles + 7) / 8;                                // 8 waves per block

    // degree + symmetric norm
    init_deg_kernel<<<gN, TB, 0, stream>>>(dinv);
    deg_edges_kernel<<<gE, TB, 0, stream>>>(ei, dinv);
    rsqrt_kernel<<<gN, TB, 0, stream>>>(dinv);

    // layer 1
    xw1_kernel<<<gN16, TB, 0, stream>>>(x, W1, B1);
    selfloop_kernel<<<gN16, TB, 0, stream>>>(B1, dinv, B2);
    edge_scatter_kernel<<<gE16, TB, 0, stream>>>(ei, dinv, B1, B2);
    bias_relu_kernel<<<gN16, TB, 0, stream>>>(B2, b1, B2);   // h1 in B2

    // layer 2
    gemm64_wmma_kernel<<<gGemm, TB, 0, stream>>>(B2, W2, B1); // xw2 in B1
    selfloop_kernel<<<gN16, TB, 0, stream>>>(B1, dinv, B2);
    edge_scatter_kernel<<<gE16, TB, 0, stream>>>(ei, dinv, B1, B2);
    bias_relu_kernel<<<gN16, TB, 0, stream>>>(B2, b2, B2);   // h2 in B2

    // heads
    logits_kernel<<<gN32, TB, 0, stream>>>(B2, Wp, bp, msk, logits);
    pool_zero_kernel<<<(N_GRAPHS * 64 + TB - 1) / TB, TB, 0, stream>>>(sums, cnts);
    pool_accum_kernel<<<gN16, TB, 0, stream>>>(B2, bat, sums, cnts);
    value_kernel<<<(N_GRAPHS * 32 + TB - 1) / TB, TB, 0, stream>>>(sums, cnts, Wv, bv, vout);
}